// GraphConvSparse_22574348108057
// MI455X (gfx1250) — compile-verified
//
#include <hip/hip_runtime.h>

// ---------------------------------------------------------------------------
// GCN layer on MI455X (gfx1250, wave32):
//   h = x @ W          (fp32 WMMA 16x16x4, LDS-staged W in K-pair layout)
//   out = relu(segment_sum(val[e] * h[col[e]] -> row[e]))
// ---------------------------------------------------------------------------

typedef float v2f __attribute__((ext_vector_type(2)));
typedef float v8f __attribute__((ext_vector_type(8)));

#define N_NODES 50000
#define D_IN    128
#define D_OUT   128
#define N_EDGES 800000

#define M_TILES   (N_NODES / 16)   // 3125 row tiles of 16
#define N_TILES   (D_OUT / 16)     // 8 col tiles of 16
#define WAVES_PER_BLOCK 8

// K-pair-interleaved W layout: sW2[(k/2)*PAIR_STRIDE + n*2 + (k&1)] = W[k][n].
// A B-fragment {W[k][n], W[k+1][n]} is then one contiguous 8B -> ds_load_b64.
// PAIR_STRIDE = 256 + 32 pad: 288 % 64 == 32, so the two half-waves (pair p
// vs p+1) hit disjoint 32-bank sets; within a half-wave lanes hit banks
// {2hl, 2hl+1} -> conflict-free.
#define PAIR_STRIDE 288
#define LDS_FLOATS  ((D_IN / 2) * PAIR_STRIDE)   // 64*288 = 18432 fl = 73,728 B

// --------------------------- zero d_out ------------------------------------
__global__ __launch_bounds__(256)
void gcn_zero_kernel(float4* __restrict__ out, int n4) {
    int i = blockIdx.x * 256 + threadIdx.x;
    if (i < n4) out[i] = make_float4(0.f, 0.f, 0.f, 0.f);
}

// --------------------------- GEMM: h = x @ W -------------------------------
// One wave computes a full 16x128 output row-block (8 x v8f accumulators),
// so x is streamed exactly once from HBM. W lives in LDS for the whole block.
__global__ __launch_bounds__(256)
void gcn_gemm_wmma_kernel(const float* __restrict__ x,
                          const float* __restrict__ w,
                          float* __restrict__ h) {
    __shared__ float sW2[LDS_FLOATS];

    const int tid = threadIdx.x;

    // Cooperative stage of W[128][128] -> LDS in K-pair layout.
    for (int idx = tid; idx < D_IN * (D_OUT / 4); idx += 256) {
        const int k  = idx >> 5;            // 0..127
        const int c4 = (idx & 31) << 2;     // 0,4,...,124
        const float4 v = *(const float4*)(w + k * D_OUT + c4);
        float* dst = &sW2[(k >> 1) * PAIR_STRIDE + c4 * 2 + (k & 1)];
        dst[0] = v.x; dst[2] = v.y; dst[4] = v.z; dst[6] = v.w;
    }
    __syncthreads();

    const int wave = tid >> 5;
    const int lane = tid & 31;
    const int tileM = blockIdx.x * WAVES_PER_BLOCK + wave;
    if (tileM >= M_TILES) return;

    // ISA A-layout for 32-bit 16x4: lanes 0-15 -> {K0,K1}, lanes 16-31 -> {K2,K3}
    const int halfLane = lane & 15;
    const int kHi      = (lane >> 4) << 1;          // 0 or 2
    const float* aRow  = x + (size_t)(tileM * 16 + halfLane) * D_IN;

    // Per-lane base into the pair rows this half-wave consumes.
    const float* bBase = &sW2[(lane >> 4) * PAIR_STRIDE + halfLane * 2];

    v8f acc[N_TILES];
    #pragma unroll
    for (int n = 0; n < N_TILES; ++n) acc[n] = (v8f)0.0f;

    #pragma unroll 4
    for (int k0 = 0; k0 < D_IN; k0 += 4) {
        // A fragment: two consecutive fp32 along K (8B aligned).
        const v2f a = *(const v2f*)(aRow + k0 + kHi);

        const float* bRow = bBase + (k0 >> 1) * PAIR_STRIDE;
        #pragma unroll
        for (int n = 0; n < N_TILES; ++n) {
            // Single ds_load_b64: {W[keven][col], W[keven+1][col]}.
            const v2f b = *(const v2f*)(bRow + n * 32);
            acc[n] = __builtin_amdgcn_wmma_f32_16x16x4_f32(
                /*neg_a=*/false, a, /*neg_b=*/false, b,
                /*c_mod=*/(short)0, acc[n],
                /*reuse_a=*/false, /*reuse_b=*/false);
        }
    }

    // C/D layout: VGPR v -> M = v (lanes 0-15) / v+8 (lanes 16-31), N = halfLane.
    const int mBase = tileM * 16 + ((lane >> 4) << 3);
    #pragma unroll
    for (int n = 0; n < N_TILES; ++n) {
        float* outRowBase = h + (size_t)mBase * D_OUT + n * 16 + halfLane;
        #pragma unroll
        for (int v = 0; v < 8; ++v)
            outRowBase[(size_t)v * D_OUT] = acc[n][v];
    }
}

// ------------------- SpMM scatter: out[row] += val * h[col] ----------------
// Wave per edge; lane owns a float4 slice of the 128-wide feature row.
// h and out are L2-resident (25.6 MB each << 192 MB), so the 4 fp32 atomics
// per lane resolve in the L2 atomic units.
__global__ __launch_bounds__(256)
void gcn_spmm_scatter_kernel(const float* __restrict__ h,
                             const int* __restrict__ edge_row,
                             const int* __restrict__ edge_col,
                             const float* __restrict__ edge_val,
                             float* __restrict__ out) {
    const int wave = threadIdx.x >> 5;
    const int lane = threadIdx.x & 31;
    const int e = blockIdx.x * WAVES_PER_BLOCK + wave;
    if (e >= N_EDGES) return;

    const int   r = edge_row[e];
    const int   c = edge_col[e];
    const float v = edge_val[e];

    const float4 hv = *(const float4*)(h + (size_t)c * D_OUT + lane * 4);
    float* o = out + (size_t)r * D_OUT + lane * 4;

    __hip_atomic_fetch_add(o + 0, v * hv.x, __ATOMIC_RELAXED, __HIP_MEMORY_SCOPE_AGENT);
    __hip_atomic_fetch_add(o + 1, v * hv.y, __ATOMIC_RELAXED, __HIP_MEMORY_SCOPE_AGENT);
    __hip_atomic_fetch_add(o + 2, v * hv.z, __ATOMIC_RELAXED, __HIP_MEMORY_SCOPE_AGENT);
    __hip_atomic_fetch_add(o + 3, v * hv.w, __ATOMIC_RELAXED, __HIP_MEMORY_SCOPE_AGENT);
}

// --------------------------- ReLU in place ---------------------------------
__global__ __launch_bounds__(256)
void gcn_relu_kernel(float4* __restrict__ out, int n4) {
    int i = blockIdx.x * 256 + threadIdx.x;
    if (i < n4) {
        float4 v = out[i];
        v.x = fmaxf(v.x, 0.f);
        v.y = fmaxf(v.y, 0.f);
        v.z = fmaxf(v.z, 0.f);
        v.w = fmaxf(v.w, 0.f);
        out[i] = v;
    }
}

// ---------------------------------------------------------------------------
extern "C" void kernel_launch(void* const* d_in, const int* in_sizes, int n_in,
                              void* d_out, int out_size, void* d_ws, size_t ws_size,
                              hipStream_t stream) {
    const float* x        = (const float*)d_in[0];
    const float* weight   = (const float*)d_in[1];
    const int*   edge_row = (const int*)d_in[2];
    const int*   edge_col = (const int*)d_in[3];
    const float* edge_val = (const float*)d_in[4];
    float*       out      = (float*)d_out;
    float*       h        = (float*)d_ws;   // 50000*128*4 = 25.6 MB scratch

    const int total  = N_NODES * D_OUT;     // 6.4M
    const int n4     = total / 4;

    // 1) zero accumulation buffer (d_out is poisoned by the harness)
    gcn_zero_kernel<<<(n4 + 255) / 256, 256, 0, stream>>>((float4*)out, n4);

    // 2) h = x @ W  (fp32 WMMA)
    const int gemmBlocks = (M_TILES + WAVES_PER_BLOCK - 1) / WAVES_PER_BLOCK; // 391
    gcn_gemm_wmma_kernel<<<gemmBlocks, 256, 0, stream>>>(x, weight, h);

    // 3) scatter-add messages
    const int edgeBlocks = (N_EDGES + WAVES_PER_BLOCK - 1) / WAVES_PER_BLOCK; // 100000
    gcn_spmm_scatter_kernel<<<edgeBlocks, 256, 0, stream>>>(h, edge_row, edge_col,
                                                           edge_val, out);

    // 4) ReLU
    gcn_relu_kernel<<<(n4 + 255) / 256, 256, 0, stream>>>((float4*)out, n4);
}